// DeepGAT_19284403159298
// MI455X (gfx1250) — compile-verified
//
#include <hip/hip_runtime.h>
#include <hip/hip_bf16.h>

#define N_NODES 50000
#define D_FEAT 128
#define HID 64
#define HEADS 2
#define N_EDGES 800000
#define N_GRAPHS 64
#define N_OUT 10
#define NEG_SLOPE 0.2f

typedef float v2f __attribute__((ext_vector_type(2)));
typedef float v8f __attribute__((ext_vector_type(8)));

// ---------------------------------------------------------------------------
// GEMM: Hout[n,128] = X[n,128] @ W[128,128], fp32 WMMA 16x16x4.
// One wave = one 16x16 output tile; 8 waves/block = one 16-row slab x 128 cols.
// A frag (16x4 f32): lanes 0-15 row M=l hold K=0,1; lanes 16-31 row M=l hold K=2,3.
// B frag (4x16 f32): lanes 0-15 col N=l hold K=0,1; lanes 16-31 col N=l hold K=2,3.
// C/D: VGPR r -> (M=r, N=lane) lanes 0-15, (M=r+8, N=lane-16) lanes 16-31.
// ---------------------------------------------------------------------------
__global__ __launch_bounds__(256) void gemm_wmma_f32_kernel(
    const float* __restrict__ X, const float* __restrict__ W,
    float* __restrict__ Hout, int nrows) {
  const int waveId = threadIdx.x >> 5;
  const int lane   = threadIdx.x & 31;
  const int rt = blockIdx.x;      // row tile (16 rows)
  const int ct = waveId;          // col tile (0..7)
  if (rt * 16 >= nrows) return;   // uniform per wave (EXEC stays all-ones)

  const int half = lane >> 4;     // 0: K=0,1   1: K=2,3
  const int l    = lane & 15;
  const float* __restrict__ xrow = X + (size_t)(rt * 16 + l) * D_FEAT;
  const int col = ct * 16 + l;

  v8f acc = {};
#pragma unroll
  for (int k = 0; k < D_FEAT; k += 4) {
    const int kk = k + 2 * half;
    v2f a, b;
    a.x = xrow[kk];
    a.y = xrow[kk + 1];
    b.x = W[(size_t)kk * D_FEAT + col];
    b.y = W[(size_t)(kk + 1) * D_FEAT + col];
    acc = __builtin_amdgcn_wmma_f32_16x16x4_f32(
        /*neg_a=*/false, a, /*neg_b=*/false, b,
        /*c_mod=*/(short)0, acc, /*reuse_a=*/false, /*reuse_b=*/false);
  }

  float* __restrict__ orow = Hout + (size_t)(rt * 16 + 8 * half) * D_FEAT + ct * 16 + l;
#pragma unroll
  for (int r = 0; r < 8; ++r) orow[(size_t)r * D_FEAT] = acc[r];
}

// ---------------------------------------------------------------------------
// Per-node attention projections: als[n,h] = sum_c h[n,h,c]*a_src[h,c], same ald.
// ---------------------------------------------------------------------------
__global__ __launch_bounds__(256) void att_scores_kernel(
    const float* __restrict__ h, const float* __restrict__ asrc,
    const float* __restrict__ adst, float* __restrict__ als,
    float* __restrict__ ald, int n) {
  int node = blockIdx.x * blockDim.x + threadIdx.x;
  if (node >= n) return;
  const float* __restrict__ hr = h + (size_t)node * D_FEAT;
  float s0 = 0.f, s1 = 0.f, d0 = 0.f, d1 = 0.f;
#pragma unroll 4
  for (int c = 0; c < HID; ++c) {
    float h0 = hr[c], h1 = hr[HID + c];
    s0 += h0 * asrc[c];       s1 += h1 * asrc[HID + c];
    d0 += h0 * adst[c];       d1 += h1 * adst[HID + c];
  }
  als[node * 2 + 0] = s0;  als[node * 2 + 1] = s1;
  ald[node * 2 + 0] = d0;  ald[node * 2 + 1] = d1;
}

// Monotone float<->uint encoding so atomicMax(uint) implements float max.
__device__ __forceinline__ unsigned fenc(float f) {
  unsigned u = __float_as_uint(f);
  return (u & 0x80000000u) ? ~u : (u | 0x80000000u);
}
__device__ __forceinline__ float fdec(unsigned u) {
  return (u & 0x80000000u) ? __uint_as_float(u ^ 0x80000000u)
                           : __uint_as_float(~u);
}

__device__ __forceinline__ void edge_nodes(const long long* __restrict__ ei,
                                           int e, long long& s, long long& d) {
  if (e < N_EDGES) { s = ei[e]; d = ei[(size_t)N_EDGES + e]; }
  else             { s = e - N_EDGES; d = s; }   // appended self-loop
}

__device__ __forceinline__ float lrelu(float x) {
  return x > 0.f ? x : NEG_SLOPE * x;
}

// Pass 1: segment max of leaky_relu(als[src]+ald[dst]) into m_enc[dst].
__global__ __launch_bounds__(256) void edge_max_kernel(
    const long long* __restrict__ ei, const float* __restrict__ als,
    const float* __restrict__ ald, unsigned* __restrict__ m_enc, int total) {
  int e = blockIdx.x * blockDim.x + threadIdx.x;
  if (e >= total) return;
  long long s, d;
  edge_nodes(ei, e, s, d);
#pragma unroll
  for (int hh = 0; hh < HEADS; ++hh) {
    float sc = lrelu(als[s * 2 + hh] + ald[d * 2 + hh]);
    atomicMax(&m_enc[d * 2 + hh], fenc(sc));
  }
}

// Pass 2: denom[dst] += exp(score - m[dst]).
__global__ __launch_bounds__(256) void edge_denom_kernel(
    const long long* __restrict__ ei, const float* __restrict__ als,
    const float* __restrict__ ald, const unsigned* __restrict__ m_enc,
    float* __restrict__ denom, int total) {
  int e = blockIdx.x * blockDim.x + threadIdx.x;
  if (e >= total) return;
  long long s, d;
  edge_nodes(ei, e, s, d);
#pragma unroll
  for (int hh = 0; hh < HEADS; ++hh) {
    float sc = lrelu(als[s * 2 + hh] + ald[d * 2 + hh]);
    float ex = expf(sc - fdec(m_enc[d * 2 + hh]));
    atomicAdd(&denom[d * 2 + hh], ex);
  }
}

// Pass 3: one WAVE per edge; recompute alpha, scatter 128 floats coalesced.
__global__ __launch_bounds__(256) void edge_aggr_kernel(
    const long long* __restrict__ ei, const float* __restrict__ als,
    const float* __restrict__ ald, const unsigned* __restrict__ m_enc,
    const float* __restrict__ denom, const float* __restrict__ h,
    float* __restrict__ agg, int total) {
  int wid  = (int)((blockIdx.x * (size_t)blockDim.x + threadIdx.x) >> 5);
  int lane = threadIdx.x & 31;
  if (wid >= total) return;
  long long s, d;
  edge_nodes(ei, wid, s, d);
  float alpha[HEADS];
#pragma unroll
  for (int hh = 0; hh < HEADS; ++hh) {
    float sc = lrelu(als[s * 2 + hh] + ald[d * 2 + hh]);
    float ex = expf(sc - fdec(m_enc[d * 2 + hh]));
    alpha[hh] = ex / denom[d * 2 + hh];
  }
  const float* __restrict__ hs = h + (size_t)s * D_FEAT;
  float* __restrict__ od = agg + (size_t)d * D_FEAT;
#pragma unroll
  for (int t = 0; t < 4; ++t) {
    int f = lane + t * 32;                 // 0..127, contiguous across lanes
    atomicAdd(&od[f], alpha[f >> 6] * hs[f]);
  }
}

// out = elu(agg + bias), in place.
__global__ __launch_bounds__(256) void bias_elu_kernel(
    float* __restrict__ agg, const float* __restrict__ b, size_t count) {
  size_t t = blockIdx.x * (size_t)blockDim.x + threadIdx.x;
  if (t >= count) return;
  float v = agg[t] + b[t & (D_FEAT - 1)];
  agg[t] = v > 0.f ? v : expm1f(v);
}

// Global mean pool: accumulate sums/counts per graph.
__global__ __launch_bounds__(256) void pool_sum_kernel(
    const float* __restrict__ feat, const long long* __restrict__ batch,
    float* __restrict__ sums, float* __restrict__ cnt, int n) {
  size_t t = blockIdx.x * (size_t)blockDim.x + threadIdx.x;
  if (t >= (size_t)n * D_FEAT) return;
  int node = (int)(t >> 7), c = (int)(t & (D_FEAT - 1));
  long long g = batch[node];
  atomicAdd(&sums[g * D_FEAT + c], feat[t]);
  if (c == 0) atomicAdd(&cnt[g], 1.0f);
}

// z = relu(pooled @ lin1_w + lin1_b)   [G x 64]
__global__ __launch_bounds__(256) void head_lin1_kernel(
    const float* __restrict__ sums, const float* __restrict__ cnt,
    const float* __restrict__ w, const float* __restrict__ b,
    float* __restrict__ z) {
  int t = blockIdx.x * blockDim.x + threadIdx.x;
  if (t >= N_GRAPHS * HID) return;
  int g = t >> 6, j = t & 63;
  float c = cnt[g];
  float inv = 1.0f / (c > 1.f ? c : 1.f);
  float acc = 0.f;
#pragma unroll 4
  for (int k = 0; k < D_FEAT; ++k)
    acc += (sums[g * D_FEAT + k] * inv) * w[k * HID + j];
  acc += b[j];
  z[t] = acc > 0.f ? acc : 0.f;
}

// out = z @ lin2_w + lin2_b   [G x 10]
__global__ __launch_bounds__(256) void head_lin2_kernel(
    const float* __restrict__ z, const float* __restrict__ w,
    const float* __restrict__ b, float* __restrict__ out) {
  int t = blockIdx.x * blockDim.x + threadIdx.x;
  if (t >= N_GRAPHS * N_OUT) return;
  int g = t / N_OUT, o = t % N_OUT;
  float acc = b[o];
#pragma unroll 8
  for (int k = 0; k < HID; ++k) acc += z[g * HID + k] * w[k * N_OUT + o];
  out[t] = acc;
}

__global__ __launch_bounds__(256) void fill_u32_kernel(
    unsigned* __restrict__ p, unsigned v, size_t count) {
  size_t t = blockIdx.x * (size_t)blockDim.x + threadIdx.x;
  if (t < count) p[t] = v;
}

static inline unsigned gridFor(size_t n, unsigned block) {
  return (unsigned)((n + block - 1) / block);
}

extern "C" void kernel_launch(void* const* d_in, const int* in_sizes, int n_in,
                              void* d_out, int out_size, void* d_ws, size_t ws_size,
                              hipStream_t stream) {
  const float*     x     = (const float*)d_in[0];
  const long long* ei    = (const long long*)d_in[1];   // [2, E] int64
  const long long* batch = (const long long*)d_in[2];   // [N]   int64
  const float* Wm[3]  = {(const float*)d_in[3],  (const float*)d_in[7],  (const float*)d_in[11]};
  const float* As[3]  = {(const float*)d_in[4],  (const float*)d_in[8],  (const float*)d_in[12]};
  const float* Ad[3]  = {(const float*)d_in[5],  (const float*)d_in[9],  (const float*)d_in[13]};
  const float* Bs[3]  = {(const float*)d_in[6],  (const float*)d_in[10], (const float*)d_in[14]};
  const float* lin1_w = (const float*)d_in[15];
  const float* lin1_b = (const float*)d_in[16];
  const float* lin2_w = (const float*)d_in[17];
  const float* lin2_b = (const float*)d_in[18];
  float* out = (float*)d_out;

  const size_t nfeat = (size_t)N_NODES * D_FEAT;
  char* ws = (char*)d_ws;
  float*    bufA  = (float*)ws;    ws += nfeat * sizeof(float);           // agg / features
  float*    bufB  = (float*)ws;    ws += nfeat * sizeof(float);           // gemm output h
  float*    als   = (float*)ws;    ws += (size_t)N_NODES * 2 * sizeof(float);
  float*    ald   = (float*)ws;    ws += (size_t)N_NODES * 2 * sizeof(float);
  unsigned* m_enc = (unsigned*)ws; ws += (size_t)N_NODES * 2 * sizeof(unsigned);
  float*    denom = (float*)ws;    ws += (size_t)N_NODES * 2 * sizeof(float);
  float*    sums  = (float*)ws;    ws += (size_t)N_GRAPHS * D_FEAT * sizeof(float);
  float*    cnt   = (float*)ws;    ws += (size_t)N_GRAPHS * sizeof(float);
  float*    zbuf  = (float*)ws;    ws += (size_t)N_GRAPHS * HID * sizeof(float);

  const int total = N_EDGES + N_NODES;   // edges + self-loops
  const unsigned BLK = 256;

  const float* feat = x;
  for (int layer = 0; layer < 3; ++layer) {
    // h = feat @ W  (fp32 WMMA)
    gemm_wmma_f32_kernel<<<N_NODES / 16, BLK, 0, stream>>>(feat, Wm[layer], bufB, N_NODES);
    // per-node attention projections
    att_scores_kernel<<<gridFor(N_NODES, BLK), BLK, 0, stream>>>(
        bufB, As[layer], Ad[layer], als, ald, N_NODES);
    // zero accumulators (m_enc==0 is below enc(-inf); every dst has a self-loop)
    fill_u32_kernel<<<gridFor((size_t)N_NODES * 2, BLK), BLK, 0, stream>>>(m_enc, 0u, (size_t)N_NODES * 2);
    fill_u32_kernel<<<gridFor((size_t)N_NODES * 2, BLK), BLK, 0, stream>>>((unsigned*)denom, 0u, (size_t)N_NODES * 2);
    fill_u32_kernel<<<gridFor(nfeat, BLK), BLK, 0, stream>>>((unsigned*)bufA, 0u, nfeat);
    // segment softmax + aggregate
    edge_max_kernel<<<gridFor(total, BLK), BLK, 0, stream>>>(ei, als, ald, m_enc, total);
    edge_denom_kernel<<<gridFor(total, BLK), BLK, 0, stream>>>(ei, als, ald, m_enc, denom, total);
    edge_aggr_kernel<<<gridFor((size_t)total * 32, BLK), BLK, 0, stream>>>(
        ei, als, ald, m_enc, denom, bufB, bufA, total);
    // + bias, ELU (in place)
    bias_elu_kernel<<<gridFor(nfeat, BLK), BLK, 0, stream>>>(bufA, Bs[layer], nfeat);
    feat = bufA;
  }

  // global mean pool + MLP head
  fill_u32_kernel<<<gridFor((size_t)N_GRAPHS * D_FEAT, BLK), BLK, 0, stream>>>((unsigned*)sums, 0u, (size_t)N_GRAPHS * D_FEAT);
  fill_u32_kernel<<<1, BLK, 0, stream>>>((unsigned*)cnt, 0u, (size_t)N_GRAPHS);
  pool_sum_kernel<<<gridFor(nfeat, BLK), BLK, 0, stream>>>(bufA, batch, sums, cnt, N_NODES);
  head_lin1_kernel<<<gridFor(N_GRAPHS * HID, BLK), BLK, 0, stream>>>(sums, cnt, lin1_w, lin1_b, zbuf);
  head_lin2_kernel<<<gridFor(N_GRAPHS * N_OUT, BLK), BLK, 0, stream>>>(zbuf, lin2_w, lin2_b, out);
}